// HashEncoding_59493886984431
// MI455X (gfx1250) — compile-verified
//
#include <hip/hip_runtime.h>
#include <math.h>
#include <float.h>

#define N_LEVELS 16
#define LOG2_HASH 19
#define HASH_SIZE (1u << LOG2_HASH)
#define HASH_MASK (HASH_SIZE - 1u)

typedef float v2f __attribute__((ext_vector_type(2)));
typedef float v4f __attribute__((ext_vector_type(4)));

// ---- exact float min/max via order-preserving uint encoding ----
__device__ __forceinline__ unsigned encodeOrd(float f) {
    unsigned u = __float_as_uint(f);
    return (u & 0x80000000u) ? ~u : (u | 0x80000000u);
}
__device__ __forceinline__ float decodeOrd(unsigned u) {
    unsigned b = (u & 0x80000000u) ? (u ^ 0x80000000u) : ~u;
    return __uint_as_float(b);
}

// ws layout: ws[0..2] = encoded min(x,y,z), ws[3..5] = encoded max(x,y,z)
__global__ void he_init_kernel(unsigned* __restrict__ ws) {
    if (threadIdx.x == 0) {
        ws[0] = 0xFFFFFFFFu; ws[1] = 0xFFFFFFFFu; ws[2] = 0xFFFFFFFFu;  // atomicMin targets
        ws[3] = 0u;          ws[4] = 0u;          ws[5] = 0u;           // atomicMax targets
    }
}

__global__ __launch_bounds__(256)
void he_minmax_kernel(const float* __restrict__ x, int n, unsigned* __restrict__ ws) {
    float mn[3] = { FLT_MAX,  FLT_MAX,  FLT_MAX};
    float mx[3] = {-FLT_MAX, -FLT_MAX, -FLT_MAX};

    for (int i = blockIdx.x * blockDim.x + threadIdx.x; i < n; i += gridDim.x * blockDim.x) {
        #pragma unroll
        for (int d = 0; d < 3; ++d) {
            float v = x[(size_t)i * 3 + d];
            mn[d] = fminf(mn[d], v);
            mx[d] = fmaxf(mx[d], v);
        }
    }

    // wave32 butterfly reduction (CDNA5 is wave32-only)
    #pragma unroll
    for (int off = 16; off > 0; off >>= 1) {
        #pragma unroll
        for (int d = 0; d < 3; ++d) {
            mn[d] = fminf(mn[d], __shfl_xor(mn[d], off, 32));
            mx[d] = fmaxf(mx[d], __shfl_xor(mx[d], off, 32));
        }
    }

    __shared__ float smn[8][3];
    __shared__ float smx[8][3];
    const int lane = threadIdx.x & 31;
    const int wave = threadIdx.x >> 5;
    if (lane == 0) {
        #pragma unroll
        for (int d = 0; d < 3; ++d) { smn[wave][d] = mn[d]; smx[wave][d] = mx[d]; }
    }
    __syncthreads();

    if (threadIdx.x == 0) {
        const int nw = (int)(blockDim.x >> 5);
        #pragma unroll
        for (int d = 0; d < 3; ++d) {
            float bmn = smn[0][d], bmx = smx[0][d];
            for (int w = 1; w < nw; ++w) {
                bmn = fminf(bmn, smn[w][d]);
                bmx = fmaxf(bmx, smx[w][d]);
            }
            atomicMin(&ws[d],     encodeOrd(bmn));
            atomicMax(&ws[3 + d], encodeOrd(bmx));
        }
    }
}

struct ResPack { float res[N_LEVELS]; };

// One hash-grid level: 8-corner gather + trilinear blend.
// Fully inlined; the 8 b64 gathers are independent so the scheduler can put
// them all in flight before the first dependent FMA.
__device__ __forceinline__ v2f he_level(const v2f* __restrict__ t, float r,
                                        float sx, float sy, float sz) {
    const float gx = sx * r, gy = sy * r, gz = sz * r;
    const float fx0 = floorf(gx), fy0 = floorf(gy), fz0 = floorf(gz);
    const float fx = gx - fx0, fy = gy - fy0, fz = gz - fz0;

    const unsigned cx = (unsigned)fx0;
    const unsigned cy = (unsigned)fy0;
    const unsigned cz = (unsigned)fz0;

    // hash components: prime[0]=1, prime[1]=2654435761, prime[2]=805459861
    const unsigned hx0 = cx;
    const unsigned hx1 = cx + 1u;
    const unsigned hy0 = cy * 2654435761u;
    const unsigned hy1 = (cy + 1u) * 2654435761u;
    const unsigned hz0 = cz * 805459861u;
    const unsigned hz1 = (cz + 1u) * 805459861u;

    const float wx0 = 1.0f - fx, wx1 = fx;
    const float wy0 = 1.0f - fy, wy1 = fy;
    const float wz0 = 1.0f - fz, wz1 = fz;

    float ax = 0.0f, ay = 0.0f;
    // corner order: dz outer, dy middle, dx inner -> c bit0=dx, bit1=dy, bit2=dz
    #pragma unroll
    for (int c = 0; c < 8; ++c) {
        const unsigned h = ((c & 1) ? hx1 : hx0)
                         ^ ((c & 2) ? hy1 : hy0)
                         ^ ((c & 4) ? hz1 : hz0);
        const float w = ((c & 1) ? wx1 : wx0)
                      * ((c & 2) ? wy1 : wy0)
                      * ((c & 4) ? wz1 : wz0);
        const v2f f = t[h & HASH_MASK];      // global_load_b64, L2-resident table
        ax = fmaf(w, f.x, ax);
        ay = fmaf(w, f.y, ay);
    }
    v2f o; o.x = ax; o.y = ay;
    return o;
}

__global__ __launch_bounds__(256)
void he_encode_kernel(const float* __restrict__ x,
                      const float* __restrict__ tables,
                      float* __restrict__ out,
                      const unsigned* __restrict__ ws,
                      int n, ResPack rp) {
    const int i = blockIdx.x * blockDim.x + threadIdx.x;
    if (i >= n) return;

    // per-dim min/max (uniform; scalar-cached loads)
    const float mn0 = decodeOrd(ws[0]), mn1 = decodeOrd(ws[1]), mn2 = decodeOrd(ws[2]);
    const float mx0 = decodeOrd(ws[3]), mx1 = decodeOrd(ws[4]), mx2 = decodeOrd(ws[5]);

    const float px = x[(size_t)i * 3 + 0];
    const float py = x[(size_t)i * 3 + 1];
    const float pz = x[(size_t)i * 3 + 2];

    const float sx = (px - mn0) / (mx0 - mn0 + 1e-8f);
    const float sy = (py - mn1) / (mx1 - mn1 + 1e-8f);
    const float sz = (pz - mn2) / (mx2 - mn2 + 1e-8f);

    const v2f* __restrict__ tb = (const v2f*)tables;
    v4f* __restrict__ outv = (v4f*)out;   // out row i = 8 x v4f

    // Two levels per iteration: 16 independent gathers in flight per wave,
    // and one 16B non-temporal store (b128, th:NT) per pair so the 128MB
    // streamed output doesn't evict the 64MB of L2-resident tables.
    for (int l = 0; l < N_LEVELS; l += 2) {
        const v2f o0 = he_level(tb + (size_t)l       * HASH_SIZE, rp.res[l],     sx, sy, sz);
        const v2f o1 = he_level(tb + (size_t)(l + 1) * HASH_SIZE, rp.res[l + 1], sx, sy, sz);
        v4f o; o.x = o0.x; o.y = o0.y; o.z = o1.x; o.w = o1.y;
        __builtin_nontemporal_store(o, &outv[(size_t)i * (N_LEVELS / 2) + (l >> 1)]);
    }
}

extern "C" void kernel_launch(void* const* d_in, const int* in_sizes, int n_in,
                              void* d_out, int out_size, void* d_ws, size_t ws_size,
                              hipStream_t stream) {
    const float* x      = (const float*)d_in[0];   // [N,3] f32
    const float* tables = (const float*)d_in[1];   // [16, 2^19, 2] f32
    float* out          = (float*)d_out;           // [N,32] f32
    unsigned* ws        = (unsigned*)d_ws;

    const int n = in_sizes[0] / 3;

    // resolutions: int(16 * growth^l), growth = exp((ln512-ln16)/15), double math
    // to match numpy's computation (incl. the truncation-sensitive levels).
    ResPack rp;
    const double growth = exp((log(512.0) - log(16.0)) / 15.0);
    for (int l = 0; l < N_LEVELS; ++l)
        rp.res[l] = (float)(int)(16.0 * pow(growth, (double)l));

    he_init_kernel<<<1, 64, 0, stream>>>(ws);

    int rblocks = (n + 255) / 256;
    if (rblocks > 1024) rblocks = 1024;
    he_minmax_kernel<<<rblocks, 256, 0, stream>>>(x, n, ws);

    const int eblocks = (n + 255) / 256;
    he_encode_kernel<<<eblocks, 256, 0, stream>>>(x, tables, out, ws, n, rp);
}